// SoftVectorQuantizer_30459908063660
// MI455X (gfx1250) — compile-verified
//
#include <hip/hip_runtime.h>
#include <hip/hip_bf16.h>

typedef float v2f __attribute__((ext_vector_type(2)));
typedef float v8f __attribute__((ext_vector_type(8)));

#define N_E      2048
#define E_DIM    16
#define N_ROWS   32768          // 32*32*32 spatial sites
#define ROWS_WG  16             // rows (z vectors) per workgroup
#define INV_TAU  (1.0f / 0.07f)
#define INV_ENT  100.0f         // 1 / 0.01
#define ENTROPY_RATIO 0.01f

// d_out layout (tuple concatenated flat, all float32):
//   [0 .. 524288)        z_q_soft  [1,16,32,32,32]  (channel-major)
//   [524288]             vq_loss
//   [524289 .. 1048577)  z_q_hard  [1,16,32,32,32]
//   [1048577 .. 1081345) indices   [1,32,32,32]     (stored as float)
#define OFF_SOFT 0
#define OFF_LOSS 524288
#define OFF_HARD 524289
#define OFF_IDX  1048577

// LDS float offsets (dynamic shared memory)
#define S_LOG   0        // [16][2048] logits slab           : 32768 floats
#define S_ZN    32768    // [16][16]   normalized z tile     : 256
#define S_M     33024    // [16] row max (phase0: inv-norm)  : 16
#define S_S1I   33040    // [16] 1/sum exp((l-m)/tau)        : 16
#define S_S2I   33056    // [16] 1/sum exp(100(l-m))         : 16
#define S_ARG   33072    // [16] argmax (int bits)           : 16
#define S_SE    33088    // [16] per-row sample entropy      : 16
#define S_SC    33104    // [8][256] GEMM2 partial C tiles   : 2048
#define SMEM_FLOATS 35152   // -> 140608 bytes (<160KB -> 2 WGs/WGP)

__device__ __forceinline__ v8f wmma4(v2f a, v2f b, v8f c) {
    // D = A(16x4,f32) * B(4x16,f32) + C(16x16,f32)
    return __builtin_amdgcn_wmma_f32_16x16x4_f32(
        false, a, false, b, (short)0, c, false, false);
}

// -------- kernel 1: l2-normalize codebook, zero accumulators --------
__global__ __launch_bounds__(256) void vq_prep(const float* __restrict__ emb,
                                               float* __restrict__ embn,
                                               float* __restrict__ ws_avg,
                                               float* __restrict__ ws_se) {
    int i = blockIdx.x * 256 + threadIdx.x;
    if (i < N_E) {
        float v[E_DIM];
        float s = 0.f;
#pragma unroll
        for (int e = 0; e < E_DIM; ++e) { v[e] = emb[i * E_DIM + e]; s += v[e] * v[e]; }
        float inv = 1.0f / fmaxf(sqrtf(s), 1e-12f);
#pragma unroll
        for (int e = 0; e < E_DIM; ++e) embn[i * E_DIM + e] = v[e] * inv;
        ws_avg[i] = 0.0f;
    }
    if (i == 0) *ws_se = 0.0f;
}

// -------- kernel 2: fused normalize -> WMMA logits -> softmax/argmax/
//                    entropy stats -> WMMA soft lookup -> outputs --------
__global__ __launch_bounds__(256) void vq_main(const float* __restrict__ z,
                                               const float* __restrict__ embn,
                                               float* __restrict__ out,
                                               float* __restrict__ ws_avg,
                                               float* __restrict__ ws_se) {
    extern __shared__ float smem[];
    float* Llog = smem + S_LOG;
    float* Lzn  = smem + S_ZN;
    float* Lm   = smem + S_M;
    float* Ls1i = smem + S_S1I;
    float* Ls2i = smem + S_S2I;
    int*   Larg = (int*)(smem + S_ARG);
    float* Lse  = smem + S_SE;
    float* Lsc  = smem + S_SC;

    const int tid   = threadIdx.x;
    const int lane  = tid & 31;
    const int wv    = tid >> 5;          // 8 waves
    const int nbase = blockIdx.x * ROWS_WG;
    const int mrow  = lane & 15;
    const int half  = lane >> 4;

    // ---- phase 0: load z tile (channel-strided), l2-normalize rows ----
    {
        int r = tid & 15, c = tid >> 4;          // consecutive tids -> consecutive rows
        Lzn[r * 16 + c] = z[c * N_ROWS + nbase + r];
    }
    __syncthreads();
    if (tid < 16) {
        float s = 0.f;
#pragma unroll
        for (int c = 0; c < 16; ++c) { float v = Lzn[tid * 16 + c]; s += v * v; }
        Lm[tid] = 1.0f / fmaxf(sqrtf(s), 1e-12f);   // Lm reused as inv-norm here
    }
    __syncthreads();
    {
        int r = tid & 15, c = tid >> 4;
        Lzn[r * 16 + c] *= Lm[r];
    }
    __syncthreads();

    // ---- phase 1: logits = zn(16x16) @ embn^T(16x2048) via WMMA f32 K=4 ----
    v2f a1[4];
#pragma unroll
    for (int q = 0; q < 4; ++q) {
        int kb = 4 * q + 2 * half;
        a1[q].x = Lzn[mrow * 16 + kb];
        a1[q].y = Lzn[mrow * 16 + kb + 1];
    }
    for (int j = wv; j < N_E / 16; j += 8) {     // 16 N-tiles per wave
        int col = j * 16 + mrow;                 // code index for this lane's B column
        v8f acc = {0.f, 0.f, 0.f, 0.f, 0.f, 0.f, 0.f, 0.f};
#pragma unroll
        for (int q = 0; q < 4; ++q) {
            int kb = 4 * q + 2 * half;
            v2f b;
            b.x = embn[col * E_DIM + kb];
            b.y = embn[col * E_DIM + kb + 1];
            acc = wmma4(a1[q], b, acc);
        }
#pragma unroll
        for (int r = 0; r < 8; ++r) {
            int m = r + 8 * half;
            Llog[m * N_E + j * 16 + mrow] = acc[r];
        }
    }
    __syncthreads();

    // ---- phase 2: per-row stats (max/argmax, tau & entropy sums) ----
#pragma unroll
    for (int rr = 0; rr < 2; ++rr) {
        int r = wv * 2 + rr;
        const float* row = Llog + r * N_E;
        float mx = -3.402823e38f;
        int   ai = 0x7fffffff;
        for (int k = lane; k < N_E; k += 32) {
            float l = row[k];
            if (l > mx) { mx = l; ai = k; }
        }
#pragma unroll
        for (int off = 16; off > 0; off >>= 1) {
            float om = __shfl_xor(mx, off, 32);
            int   oi = __shfl_xor(ai, off, 32);
            if (om > mx || (om == mx && oi < ai)) { mx = om; ai = oi; }
        }
        float s1 = 0.f, s2 = 0.f, u2 = 0.f;
        for (int k = lane; k < N_E; k += 32) {
            float l  = row[k];
            s1 += __expf((l - mx) * INV_TAU);
            float e2 = __expf((l - mx) * INV_ENT);
            s2 += e2;
            u2 += e2 * (l * INV_ENT);
        }
#pragma unroll
        for (int off = 16; off > 0; off >>= 1) {
            s1 += __shfl_xor(s1, off, 32);
            s2 += __shfl_xor(s2, off, 32);
            u2 += __shfl_xor(u2, off, 32);
        }
        if (lane == 0) {
            Lm[r]   = mx;
            Ls1i[r] = 1.0f / s1;
            Ls2i[r] = 1.0f / s2;
            Larg[r] = ai;
            // sample entropy row: lse(x) - sum(pe * x), x = l/0.01
            Lse[r] = mx * INV_ENT + logf(s2) - u2 / s2;
            int n = nbase + r;
            out[OFF_IDX + n] = (float)ai;
#pragma unroll
            for (int e = 0; e < E_DIM; ++e)
                out[OFF_HARD + e * N_ROWS + n] = embn[ai * E_DIM + e];
        }
    }
    __syncthreads();

    // ---- phase 4: z_q_soft = softmax_tau(logits) @ embn via WMMA, K=2048 ----
    // probs (unnormalized e1) generated on the fly from LDS logits.
    {
        v8f acc = {0.f, 0.f, 0.f, 0.f, 0.f, 0.f, 0.f, 0.f};
        const float rm = Lm[mrow];
        for (int q = 0; q < 64; ++q) {           // 256 K-values per wave
            int kb = wv * 256 + q * 4 + 2 * half;
            v2f a2, b2;
            a2.x = __expf((Llog[mrow * N_E + kb]     - rm) * INV_TAU);
            a2.y = __expf((Llog[mrow * N_E + kb + 1] - rm) * INV_TAU);
            b2.x = embn[kb * E_DIM + mrow];
            b2.y = embn[(kb + 1) * E_DIM + mrow];
            acc = wmma4(a2, b2, acc);
        }
#pragma unroll
        for (int r = 0; r < 8; ++r)
            Lsc[wv * 256 + r * 32 + lane] = acc[r];
    }
    __syncthreads();

    // ---- phase 5: reduce GEMM2 partials, write soft output ----
    {
        int m = tid >> 4, e = tid & 15;
        int lidx = e + 16 * (m >> 3);
        int vr   = m & 7;
        float s = 0.f;
#pragma unroll
        for (int w = 0; w < 8; ++w) s += Lsc[w * 256 + vr * 32 + lidx];
        out[OFF_SOFT + e * N_ROWS + nbase + m] = s * Ls1i[m];
    }
    if (tid == 0) {
        float s = 0.f;
#pragma unroll
        for (int r = 0; r < ROWS_WG; ++r) s += Lse[r];
        atomicAdd(ws_se, s);
    }
    // avg_probs partial: column-parallel over codes
    for (int k = tid; k < N_E; k += 256) {
        float s = 0.f;
#pragma unroll
        for (int r = 0; r < ROWS_WG; ++r)
            s += __expf((Llog[r * N_E + k] - Lm[r]) * INV_ENT) * Ls2i[r];
        atomicAdd(ws_avg + k, s);
    }
}

// -------- kernel 3: scalar loss --------
__global__ __launch_bounds__(256) void vq_finalize(const float* __restrict__ ws_avg,
                                                   const float* __restrict__ ws_se,
                                                   float* __restrict__ out) {
    __shared__ float red[256];
    int tid = threadIdx.x;
    float t = 0.f;
    for (int k = tid; k < N_E; k += 256) {
        float a = ws_avg[k] * (1.0f / (float)N_ROWS);
        t += a * logf(a + 1e-6f);
    }
    red[tid] = t;
    __syncthreads();
    for (int off = 128; off > 0; off >>= 1) {
        if (tid < off) red[tid] += red[tid + off];
        __syncthreads();
    }
    if (tid == 0) {
        float avg_entropy    = -red[0];
        float sample_entropy = (*ws_se) * (1.0f / (float)N_ROWS);
        out[OFF_LOSS] = ENTROPY_RATIO * (sample_entropy - avg_entropy);
    }
}

extern "C" void kernel_launch(void* const* d_in, const int* in_sizes, int n_in,
                              void* d_out, int out_size, void* d_ws, size_t ws_size,
                              hipStream_t stream) {
    const float* z   = (const float*)d_in[0];   // [1,16,32,32,32]
    const float* emb = (const float*)d_in[1];   // [2048,16]
    float* out = (float*)d_out;

    float* w      = (float*)d_ws;
    float* embn   = w;               // 2048*16
    float* ws_avg = w + 32768;       // 2048
    float* ws_se  = w + 34816;       // 1

    (void)in_sizes; (void)n_in; (void)out_size; (void)ws_size;

    const size_t smem_bytes = (size_t)SMEM_FLOATS * sizeof(float);  // 140608 B
    hipFuncSetAttribute((const void*)vq_main,
                        hipFuncAttributeMaxDynamicSharedMemorySize,
                        (int)smem_bytes);

    vq_prep<<<N_E / 256, 256, 0, stream>>>(emb, embn, ws_avg, ws_se);
    vq_main<<<N_ROWS / ROWS_WG, 256, smem_bytes, stream>>>(z, embn, out, ws_avg, ws_se);
    vq_finalize<<<1, 256, 0, stream>>>(ws_avg, ws_se, out);
}